// GlobalRouters_8512625180869
// MI455X (gfx1250) — compile-verified
//
#include <hip/hip_runtime.h>
#include <math.h>

typedef __attribute__((ext_vector_type(2))) float v2f;
typedef __attribute__((ext_vector_type(8))) float v8f;
typedef __attribute__((address_space(3))) float lds_f;

#define B_N 8
#define S_N 2048
#define D_N 2048
#define DS_N 64
#define M_N (B_N * S_N)        // 16384 rows
#define NUSE 1536              // neurons actually used (kn slice unused)
#define NTILES (M_N / 16)      // 1024 row-tiles for routing
#define OUT_PER_B 1792
#define NEG_INF (-3.402823466e38f)
#define EPS_F 1e-8f

// ---------------------------------------------------------------------------
// Async global->LDS copy (CDNA5): 16B per lane, tracked by ASYNCcnt.
// VDST VGPR carries the LDS byte offset (addrspace(3) ptrtoint).
// ---------------------------------------------------------------------------
__device__ __forceinline__ void async_copy_b128(float* lds_ptr, const float* gptr) {
    unsigned loff = (unsigned)(uintptr_t)(lds_f*)lds_ptr;
    asm volatile("global_load_async_to_lds_b128 %0, %1, off"
                 :: "v"(loff), "v"(gptr) : "memory");
}

// ---------------------------------------------------------------------------
// Kernel 1: normalize neuron embeddings (first 1536 rows only). 1 wave / row.
// ---------------------------------------------------------------------------
__global__ __launch_bounds__(128) void norm_emb_k(const float* __restrict__ emb,
                                                  float* __restrict__ emb_n) {
    int wave = threadIdx.x >> 5;
    int lane = threadIdx.x & 31;
    int row  = blockIdx.x * 4 + wave;          // 384 blocks * 4 waves = 1536
    const float* src = emb + (size_t)row * DS_N + lane * 2;
    float a = src[0], b = src[1];
    float ss = a * a + b * b;
    #pragma unroll
    for (int off = 16; off > 0; off >>= 1)
        ss += __shfl_xor(ss, off, 32);
    float inv = 1.0f / sqrtf(ss);
    float* dst = emb_n + (size_t)row * DS_N + lane * 2;
    dst[0] = a * inv;
    dst[1] = b * inv;
}

// ---------------------------------------------------------------------------
// Kernel 2: h[16384,64] = x[16384,2048] @ W^T + b   (f32 WMMA 16x16x4)
// 128-row tiles. K staged in 32-wide chunks, DOUBLE-BUFFERED via
// global_load_async_to_lds_b128; s_wait_asynccnt drains the consumed buffer
// while the prefetch stays in flight. Stride 34: even => aligned ds_load_b64,
// and 34*m mod 64 distinct for m=0..15 => conflict-free fragment reads.
// ---------------------------------------------------------------------------
#define G2_BK 32
#define G2_STR 34
#define XS_FLOATS (128 * G2_STR)            // 4352
#define WL_FLOATS (64 * G2_STR)             // 2176
#define BUF_FLOATS (XS_FLOATS + WL_FLOATS)  // 6528 -> 2 bufs = 52224 B LDS

__device__ __forceinline__ void issue_chunk(const float* __restrict__ x,
                                            const float* __restrict__ W,
                                            int m0, int k0,
                                            float* xs, float* wl, int tid) {
    // x tile: 128 rows x 8 float4 chunks = 1024 chunks / 256 threads
    #pragma unroll
    for (int i = 0; i < 4; ++i) {
        int c = tid + i * 256;
        int r = c >> 3, kc = (c & 7) * 4;
        async_copy_b128(&xs[r * G2_STR + kc],
                        x + (size_t)(m0 + r) * D_N + k0 + kc);
    }
    // W tile: 64 rows x 8 chunks = 512 chunks / 256 threads
    #pragma unroll
    for (int i = 0; i < 2; ++i) {
        int c = tid + i * 256;
        int r = c >> 3, kc = (c & 7) * 4;
        async_copy_b128(&wl[r * G2_STR + kc],
                        W + (size_t)r * D_N + k0 + kc);
    }
}

__global__ __launch_bounds__(256) void gemm1_k(const float* __restrict__ x,
                                               const float* __restrict__ W,
                                               const float* __restrict__ bias,
                                               float* __restrict__ h) {
    __shared__ float smem[2 * BUF_FLOATS];
    const int tid  = threadIdx.x;
    const int wave = tid >> 5;
    const int lane = tid & 31;
    const int m0   = blockIdx.x * 128;

    v8f acc[4] = {};
    const int nl   = lane & 15;
    const int koff = (lane < 16) ? 0 : 2;
    const int mrow = wave * 16 + nl;

    constexpr int NIT = D_N / G2_BK;        // 64 K-chunks

    // prologue: fill buffer 0
    issue_chunk(x, W, m0, 0, smem, smem + XS_FLOATS, tid);

    for (int it = 0; it < NIT; ++it) {
        float* xs = smem + (it & 1) * BUF_FLOATS;
        float* wl = xs + XS_FLOATS;

        if (it + 1 < NIT) {
            float* xs2 = smem + ((it + 1) & 1) * BUF_FLOATS;
            issue_chunk(x, W, m0, (it + 1) * G2_BK, xs2, xs2 + XS_FLOATS, tid);
            // 12 async loads in flight; drain the 6 belonging to this chunk
            asm volatile("s_wait_asynccnt 0x6" ::: "memory");
        } else {
            asm volatile("s_wait_asynccnt 0x0" ::: "memory");
        }
        __syncthreads();                    // chunk visible to all waves

        #pragma unroll
        for (int kk = 0; kk < G2_BK; kk += 4) {
            v2f a = *(const v2f*)&xs[mrow * G2_STR + kk + koff];
            #pragma unroll
            for (int nt = 0; nt < 4; ++nt) {
                v2f bf = *(const v2f*)&wl[(nt * 16 + nl) * G2_STR + kk + koff];
                acc[nt] = __builtin_amdgcn_wmma_f32_16x16x4_f32(
                    false, a, false, bf, (short)0, acc[nt], false, false);
            }
        }
        __syncthreads();                    // all waves done reading buffer
    }

    // store D: VGPR r -> (row r, lanes 0-15) / (row r+8, lanes 16-31)
    const int mbase = m0 + wave * 16 + ((lane < 16) ? 0 : 8);
    #pragma unroll
    for (int nt = 0; nt < 4; ++nt) {
        float bv = bias[nt * 16 + nl];
        #pragma unroll
        for (int r = 0; r < 8; ++r)
            h[(size_t)(mbase + r) * DS_N + nt * 16 + nl] = acc[nt][r] + bv;
    }
}

// ---------------------------------------------------------------------------
// Kernel 3: per 16-row tile: logits (WMMA), segment softmax, importance-
// weighted column partials -> partial[tile][1536]. Deterministic (no atomics).
// ---------------------------------------------------------------------------
#define HS_STR 66
#define LG_STR 514
__global__ __launch_bounds__(256) void route_k(const float* __restrict__ h,
                                               const float* __restrict__ embn,
                                               const float* __restrict__ imp,
                                               float* __restrict__ partial) {
    __shared__ float hs[16 * HS_STR];      // h tile 16x64 (padded stride)
    __shared__ float lg[16 * LG_STR];      // logits / exp values for a segment
    __shared__ float simp[16];
    __shared__ float rfac[16];             // importance / softmax-denominator

    const int tid  = threadIdx.x;
    const int wave = tid >> 5;
    const int lane = tid & 31;
    const int row0 = blockIdx.x * 16;      // tiles never cross a batch (2048%16==0)

    // load h tile: 16 rows x 16 float4 chunks = 256 chunks / 256 threads
    {
        int r = tid >> 4, kc = (tid & 15) * 4;
        float4 v = *(const float4*)(h + (size_t)(row0 + r) * DS_N + kc);
        float* d = &hs[r * HS_STR + kc];
        d[0] = v.x; d[1] = v.y; d[2] = v.z; d[3] = v.w;
    }
    if (tid < 16) simp[tid] = imp[row0 + tid];
    __syncthreads();

    const int nl   = lane & 15;
    const int koff = (lane < 16) ? 0 : 2;
    const int mo   = (lane < 16) ? 0 : 8;

    float* pout = partial + (size_t)blockIdx.x * NUSE;

    #pragma unroll
    for (int seg = 0; seg < 4; ++seg) {
        const int segbase = (seg == 0) ? 0 : (seg == 1) ? 512 : (seg == 2) ? 1024 : 1280;
        const int segw    = (seg < 2) ? 512 : 256;
        const int colsPW  = segw >> 3;          // columns per wave (64 or 32)
        const int ntiles  = colsPW >> 4;        // 4 or 2 WMMA n-tiles

        // --- logits via WMMA: A from LDS h-tile, B from L2-resident emb_norm
        v8f acc[4] = {};
        for (int kk = 0; kk < DS_N; kk += 4) {
            v2f a = *(const v2f*)&hs[nl * HS_STR + kk + koff];
            for (int nt = 0; nt < ntiles; ++nt) {
                int n = segbase + wave * colsPW + nt * 16 + nl;
                v2f bf = *(const v2f*)&embn[(size_t)n * DS_N + kk + koff];
                acc[nt] = __builtin_amdgcn_wmma_f32_16x16x4_f32(
                    false, a, false, bf, (short)0, acc[nt], false, false);
            }
        }
        for (int nt = 0; nt < ntiles; ++nt) {
            int col = wave * colsPW + nt * 16 + nl;   // column within segment
            #pragma unroll
            for (int r = 0; r < 8; ++r)
                lg[(mo + r) * LG_STR + col] = acc[nt][r];
        }
        __syncthreads();

        // --- softmax per row: each wave owns 2 rows
        #pragma unroll
        for (int rr = 0; rr < 2; ++rr) {
            int r = wave * 2 + rr;
            float mx = NEG_INF;
            for (int c = lane; c < segw; c += 32)
                mx = fmaxf(mx, lg[r * LG_STR + c]);
            #pragma unroll
            for (int off = 16; off > 0; off >>= 1)
                mx = fmaxf(mx, __shfl_xor(mx, off, 32));
            float s = 0.0f;
            for (int c = lane; c < segw; c += 32) {
                float e = expf(lg[r * LG_STR + c] - mx);
                lg[r * LG_STR + c] = e;
                s += e;
            }
            #pragma unroll
            for (int off = 16; off > 0; off >>= 1)
                s += __shfl_xor(s, off, 32);
            if (lane == 0) rfac[r] = simp[r] / s;
        }
        __syncthreads();

        // --- importance-weighted column sums (fixed order => deterministic)
        for (int c = tid; c < segw; c += 256) {
            float v = 0.0f;
            #pragma unroll
            for (int r = 0; r < 16; ++r)
                v += rfac[r] * lg[r * LG_STR + c];
            pout[segbase + c] = v;
        }
        __syncthreads();
    }
}

// ---------------------------------------------------------------------------
// Kernel 4: deterministic reduction of 128 tiles per batch -> dense[8,1536]
// ---------------------------------------------------------------------------
__global__ __launch_bounds__(256) void reduce_dense_k(const float* __restrict__ partial,
                                                      float* __restrict__ dense) {
    int g = blockIdx.x * 256 + threadIdx.x;          // 48 blocks -> 12288
    if (g >= B_N * NUSE) return;
    int b = g / NUSE;
    int n = g - b * NUSE;
    const float* p = partial + (size_t)(b * 128) * NUSE + n;
    float s = 0.0f;
    for (int t = 0; t < 128; ++t) s += p[(size_t)t * NUSE];
    dense[g] = s;
}

// ---------------------------------------------------------------------------
// Kernel 5: top-k sparsify + renormalize + concat. 1 wave per (batch, seg).
// ---------------------------------------------------------------------------
__global__ __launch_bounds__(32) void topk_k(const float* __restrict__ dense,
                                             float* __restrict__ out) {
    const int task = blockIdx.x;           // 32 tasks
    const int b    = task >> 2;
    const int seg  = task & 3;
    const int lane = threadIdx.x;

    const int segbase = (seg == 0) ? 0 : (seg == 1) ? 512 : (seg == 2) ? 1024 : 1280;
    const int segw    = (seg < 2) ? 512 : 256;
    const int k       = (seg < 2) ? 8 : (seg == 2) ? 4 : 6;
    const int outbase = (seg == 0) ? 0 : (seg == 1) ? 512 : (seg == 2) ? 1024 : 1536;
    const int nch     = segw >> 5;         // chunks per lane (16 or 8)

    float v[16], vc[16];
    const float* dsrc = dense + (size_t)b * NUSE + segbase;
    for (int i = 0; i < nch; ++i) {
        v[i]  = dsrc[lane + i * 32];
        vc[i] = v[i];
    }

    float sum = 0.0f;
    for (int it = 0; it < k; ++it) {
        float best = NEG_INF;
        int   bi   = 0x7fffffff;
        for (int i = 0; i < nch; ++i) {
            if (vc[i] > best) { best = vc[i]; bi = lane + i * 32; }
        }
        #pragma unroll
        for (int off = 16; off > 0; off >>= 1) {
            float ov = __shfl_xor(best, off, 32);
            int   oi = __shfl_xor(bi,   off, 32);
            if (ov > best || (ov == best && oi < bi)) { best = ov; bi = oi; }
        }
        sum += best;
        if ((bi & 31) == lane) {
            int ii = bi >> 5;
            for (int i = 0; i < 16; ++i)
                if (i == ii) vc[i] = NEG_INF;     // mark selected
        }
    }

    const float inv = 1.0f / (sum + EPS_F);
    float* ob = out + (size_t)b * OUT_PER_B;
    for (int i = 0; i < nch; ++i) {
        int idx = lane + i * 32;
        float ov = (vc[i] == NEG_INF) ? v[i] * inv : 0.0f;
        ob[outbase + idx] = ov;
        if (seg == 2) ob[1280 + idx] = ov;        // rk duplicates rq
    }
}

// ---------------------------------------------------------------------------
extern "C" void kernel_launch(void* const* d_in, const int* in_sizes, int n_in,
                              void* d_out, int out_size, void* d_ws, size_t ws_size,
                              hipStream_t stream) {
    (void)in_sizes; (void)n_in; (void)out_size; (void)ws_size;
    const float* x    = (const float*)d_in[0];
    const float* imp  = (const float*)d_in[1];
    const float* W    = (const float*)d_in[2];
    const float* bias = (const float*)d_in[3];
    const float* emb  = (const float*)d_in[4];
    float*       out  = (float*)d_out;

    // workspace layout (floats): emb_n | h | partial | dense  (~10.4 MB)
    float* emb_n   = (float*)d_ws;
    float* h       = emb_n + (size_t)NUSE * DS_N;
    float* hbuf    = h;
    float* partial = h + (size_t)M_N * DS_N;
    float* dense   = partial + (size_t)NTILES * NUSE;

    norm_emb_k    <<<NUSE / 4, 128, 0, stream>>>(emb, emb_n);
    gemm1_k       <<<M_N / 128, 256, 0, stream>>>(x, W, bias, hbuf);
    route_k       <<<NTILES, 256, 0, stream>>>(hbuf, emb_n, imp, partial);
    reduce_dense_k<<<(B_N * NUSE + 255) / 256, 256, 0, stream>>>(partial, dense);
    topk_k        <<<32, 32, 0, stream>>>(dense, out);
}